// BatchNorm2dCubic_90374701842754
// MI455X (gfx1250) — compile-verified
//
#include <hip/hip_runtime.h>

// BatchNorm2d with cubic-spline-interpolated parameters.
// apply_affine_kernel (the hot kernel, placed FIRST in the TU so the disasm
//   snippet shows it): y = fma(x, scale[c], shift[c]) with b128 non-temporal
//   loads/stores — pure HBM-bandwidth streaming, ~205 MB moved, ~8.8us floor.
// spline_params_kernel: 1 block x 256 threads (1 channel/lane) solves 4
//   natural cubic splines per channel (Thomas, size-8 tridiagonal) and writes
//   per-channel scale/shift into d_ws.

typedef float f32x4 __attribute__((ext_vector_type(4)));

#define T_KNOTS 10
#define C_CH    256
#define HW      3136      // 56*56
#define HW4     784       // HW/4
#define BN_EPS  1e-5f

__global__ void __launch_bounds__(256)
apply_affine_kernel(const f32x4* __restrict__ x,
                    const float* __restrict__ ss,
                    f32x4* __restrict__ out,
                    int n4)
{
    const int i = blockIdx.x * blockDim.x + threadIdx.x;
    if (i >= n4) return;

    const int plane = i / HW4;            // (b*C + c)
    const int c     = plane & (C_CH - 1); // C == 256, power of two

    const float scale = ss[c];            // tiny hot table: regular cached loads
    const float shift = ss[C_CH + c];

    // Streaming data: non-temporal b128 load + store (TH=NT) — each element
    // is touched exactly once; keep the 192MB L2 from churning on a
    // use-once 205MB stream.
    f32x4 v = __builtin_nontemporal_load(&x[i]);
    f32x4 r;
    r.x = fmaf(v.x, scale, shift);
    r.y = fmaf(v.y, scale, shift);
    r.z = fmaf(v.z, scale, shift);
    r.w = fmaf(v.w, scale, shift);
    __builtin_nontemporal_store(r, &out[i]);
}

__device__ __forceinline__ float spline_eval_chan(
    float tval, const float tm[T_KNOTS], const float h9[T_KNOTS - 1],
    const float y[T_KNOTS], int idx)
{
    // slopes and RHS of the natural-spline tridiagonal system
    float slopes[9];
#pragma unroll
    for (int i = 0; i < 9; ++i) slopes[i] = (y[i + 1] - y[i]) / h9[i];
    float rhs[8];
#pragma unroll
    for (int i = 0; i < 8; ++i) rhs[i] = 6.0f * (slopes[i + 1] - slopes[i]);

    // Thomas algorithm for A m_int = rhs
    // diag[i] = 2*(h[i]+h[i+1]); A[i][i+1] = h[i+1] (i<7); A[i][i-1] = h[i] (i>0)
    float cp[8], dp[8];
    float diag0 = 2.0f * (h9[0] + h9[1]);
    cp[0] = h9[1] / diag0;
    dp[0] = rhs[0] / diag0;
#pragma unroll
    for (int i = 1; i < 8; ++i) {
        float diag  = 2.0f * (h9[i] + h9[i + 1]);
        float sub   = h9[i];
        float denom = diag - sub * cp[i - 1];
        float sup   = (i < 7) ? h9[i + 1] : 0.0f;
        cp[i] = sup / denom;
        dp[i] = (rhs[i] - sub * dp[i - 1]) / denom;
    }
    float mint[8];
    mint[7] = dp[7];
#pragma unroll
    for (int i = 6; i >= 0; --i) mint[i] = dp[i] - cp[i] * mint[i + 1];

    float m[10];
    m[0] = 0.0f; m[9] = 0.0f;
#pragma unroll
    for (int i = 0; i < 8; ++i) m[i + 1] = mint[i];

    // Pick interval coefficients with compile-time indices only (selects keep
    // everything in VGPRs, no scratch).
    float a = 0.0f, b = 0.0f, c = 0.0f, d = 0.0f, u = 0.0f;
#pragma unroll
    for (int i = 0; i < 9; ++i) {
        if (i == idx) {
            a = y[i];
            b = slopes[i] - h9[i] * (2.0f * m[i] + m[i + 1]) * (1.0f / 6.0f);
            c = m[i] * 0.5f;
            d = (m[i + 1] - m[i]) / (6.0f * h9[i]);
            u = tval - tm[i];
        }
    }
    return a + u * (b + u * (c + u * d));
}

__global__ void __launch_bounds__(C_CH)
spline_params_kernel(const float* __restrict__ means,
                     const float* __restrict__ vars_,
                     const float* __restrict__ bnw,
                     const float* __restrict__ bnb,
                     const float* __restrict__ times,
                     const float* __restrict__ tptr,
                     float* __restrict__ ss)
{
    const int c = threadIdx.x;
    const float t = tptr[0];

    float tm[T_KNOTS];
#pragma unroll
    for (int i = 0; i < T_KNOTS; ++i) tm[i] = times[i];
    float h9[T_KNOTS - 1];
#pragma unroll
    for (int i = 0; i < T_KNOTS - 1; ++i) h9[i] = tm[i + 1] - tm[i];

    // idx = clip(searchsorted(times, t, 'right') - 1, 0, 8)
    int idx = 0;
#pragma unroll
    for (int i = 1; i < T_KNOTS; ++i) if (tm[i] <= t) idx = i;
    if (idx > T_KNOTS - 2) idx = T_KNOTS - 2;

    float ym[T_KNOTS], yv[T_KNOTS], yw[T_KNOTS], yb[T_KNOTS];
#pragma unroll
    for (int i = 0; i < T_KNOTS; ++i) {
        ym[i] = means[i * C_CH + c];
        yv[i] = vars_[i * C_CH + c];
        yw[i] = bnw  [i * C_CH + c];
        yb[i] = bnb  [i * C_CH + c];
    }

    const float mean   = spline_eval_chan(t, tm, h9, ym, idx);
    const float var    = spline_eval_chan(t, tm, h9, yv, idx);
    const float weight = spline_eval_chan(t, tm, h9, yw, idx);
    const float bias   = spline_eval_chan(t, tm, h9, yb, idx);

    const float inv   = rsqrtf(var + BN_EPS);
    const float scale = weight * inv;
    const float shift = bias - mean * scale;

    ss[c]        = scale;
    ss[C_CH + c] = shift;
}

extern "C" void kernel_launch(void* const* d_in, const int* in_sizes, int n_in,
                              void* d_out, int out_size, void* d_ws, size_t ws_size,
                              hipStream_t stream) {
    const float* x     = (const float*)d_in[0];
    const float* means = (const float*)d_in[1];
    const float* vars_ = (const float*)d_in[2];
    const float* bnw   = (const float*)d_in[3];
    const float* bnb   = (const float*)d_in[4];
    const float* times = (const float*)d_in[5];
    const float* t     = (const float*)d_in[6];
    float* ss  = (float*)d_ws;            // [2*C] = 2 KB scratch: scale | shift
    float* out = (float*)d_out;

    spline_params_kernel<<<1, C_CH, 0, stream>>>(means, vars_, bnw, bnb, times, t, ss);

    const int n4      = out_size / 4;     // 6,422,528 float4s
    const int threads = 256;              // 8 wave32s per block
    const int blocks  = (n4 + threads - 1) / threads;
    apply_affine_kernel<<<blocks, threads, 0, stream>>>(
        (const f32x4*)x, ss, (f32x4*)out, n4);
}